// PolySpectralConv_9174050144899
// MI455X (gfx1250) — compile-verified
//
#include <hip/hip_runtime.h>
#include <hip/hip_bf16.h>
#include <math.h>

// ---------------------------------------------------------------------------
// PolySpectralConv for MI455X (gfx1250), fp32 end-to-end.
// Dense GEMMs use V_WMMA_F32_16X16X4_F32 (wave32 WMMA). Sparse part is the
// bandwidth-bound bottleneck: gather + global_atomic_add_f32 scatter.
// ---------------------------------------------------------------------------

typedef __attribute__((ext_vector_type(2))) float v2f;
typedef __attribute__((ext_vector_type(8))) float v8f;

#define INC 128
#define OUTC 64

// ----------------- WMMA helpers (fp32 16x16x4) -----------------------------
// A (16x4, row-major source): lane L holds row m=L&15; VGPR0/1 = K pair
//   (L<16 -> K=0,1 ; L>=16 -> K=2,3)
// B (4x16, row-major source): lane L holds col n=L&15 of K rows (same split)
// C/D (16x16): VGPR i holds row i (lanes 0-15) and row i+8 (lanes 16-31)
__device__ __forceinline__ v8f wmma_f32_tile(const float* __restrict__ A, int lda,
                                             const float* __restrict__ B, int ldb,
                                             int row0, int col0, int KC, v8f acc) {
    const int lane  = threadIdx.x & 31;
    const int mn    = lane & 15;
    const int khalf = (lane >> 4) * 2;
    for (int kk = 0; kk < KC; kk += 4) {
        v2f a, b;
        const float* ap = A + (size_t)(row0 + mn) * lda + kk + khalf;
        a.x = ap[0];
        a.y = ap[1];
        b.x = B[(size_t)(kk + khalf)     * ldb + col0 + mn];
        b.y = B[(size_t)(kk + khalf + 1) * ldb + col0 + mn];
        acc = __builtin_amdgcn_wmma_f32_16x16x4_f32(false, a, false, b,
                                                    (short)0, acc, false, false);
    }
    return acc;
}

__device__ __forceinline__ void store_tile(float* __restrict__ C, int ldc,
                                           int row0, int col0, v8f acc) {
    const int lane  = threadIdx.x & 31;
    const int n     = lane & 15;
    const int mbase = (lane >> 4) * 8;
#pragma unroll
    for (int i = 0; i < 8; ++i)
        C[(size_t)(row0 + mbase + i) * ldc + col0 + n] = acc[i];
}

// ----------------- GEMM kernels --------------------------------------------
// h0 = x @ W_in   ([N,128] x [128,64])
__global__ __launch_bounds__(128) void gemm_in_kernel(const float* __restrict__ x,
                                                      const float* __restrict__ Win,
                                                      float* __restrict__ h0) {
    const int row0 = blockIdx.x * 16;
    const int col0 = (threadIdx.x >> 5) * 16;   // 4 waves -> 4 col tiles
    v8f acc = {};
    acc = wmma_f32_tile(x, INC, Win, OUTC, row0, col0, INC, acc);
    store_tile(h0, OUTC, row0, col0, acc);
}

// H = Hacc @ W_out + b_out   ([N,64] x [64,64])
__global__ __launch_bounds__(128) void gemm_out_kernel(const float* __restrict__ Hacc,
                                                       const float* __restrict__ Wout,
                                                       const float* __restrict__ bout,
                                                       float* __restrict__ H) {
    const int row0 = blockIdx.x * 16;
    const int col0 = (threadIdx.x >> 5) * 16;
    const int lane = threadIdx.x & 31;
    v8f acc = {};
    acc = wmma_f32_tile(Hacc, OUTC, Wout, OUTC, row0, col0, OUTC, acc);
    const float bv = bout[col0 + (lane & 15)];
#pragma unroll
    for (int i = 0; i < 8; ++i) acc[i] += bv;
    store_tile(H, OUTC, row0, col0, acc);
}

// g = sigmoid(H @ Wg_top + h0 @ Wg_bot + b_gate); out = g*H + (1-g)*h0
__global__ __launch_bounds__(128) void gate_kernel(const float* __restrict__ H,
                                                   const float* __restrict__ h0,
                                                   const float* __restrict__ Wg,
                                                   const float* __restrict__ bg,
                                                   float* __restrict__ out) {
    const int row0 = blockIdx.x * 16;
    const int col0 = (threadIdx.x >> 5) * 16;
    const int lane = threadIdx.x & 31;
    v8f acc = {};
    acc = wmma_f32_tile(H,  OUTC, Wg,               OUTC, row0, col0, OUTC, acc);
    acc = wmma_f32_tile(h0, OUTC, Wg + OUTC * OUTC, OUTC, row0, col0, OUTC, acc);
    const int n     = lane & 15;
    const int mbase = (lane >> 4) * 8;
    const float bv  = bg[col0 + n];
#pragma unroll
    for (int i = 0; i < 8; ++i) {
        const size_t idx = (size_t)(row0 + mbase + i) * OUTC + col0 + n;
        const float hv  = H[idx];
        const float h0v = h0[idx];
        const float g   = 1.0f / (1.0f + __expf(-(acc[i] + bv)));
        out[idx] = g * hv + (1.0f - g) * h0v;
    }
}

// ----------------- Graph kernels -------------------------------------------
__global__ void deg_kernel(float* __restrict__ deg, const float* __restrict__ w,
                           const int* __restrict__ row, int E) {
    int e = blockIdx.x * blockDim.x + threadIdx.x;
    if (e < E) atomicAdd(&deg[row[e]], w[e]);
}

__global__ void norm_kernel(float* __restrict__ nrm, const float* __restrict__ deg,
                            const float* __restrict__ w, const int* __restrict__ row,
                            const int* __restrict__ col, int E) {
    int e = blockIdx.x * blockDim.x + threadIdx.x;
    if (e < E) {
        float dr = rsqrtf(fmaxf(deg[row[e]], 1e-12f));
        float dc = rsqrtf(fmaxf(deg[col[e]], 1e-12f));
        nrm[e] = dr * w[e] * dc;
    }
}

// dst[row] += alpha * norm * src[col] ; 16 threads/edge, float4 gather
__global__ void spmm_scatter_kernel(float* __restrict__ dst, const float* __restrict__ src,
                                    const float* __restrict__ nrm,
                                    const int* __restrict__ row, const int* __restrict__ col,
                                    float alpha, int E) {
    int t = blockIdx.x * blockDim.x + threadIdx.x;
    int e = t >> 4;
    if (e >= E) return;
    int cg = (t & 15) * 4;
    const int r = row[e];
    const int c = col[e];
    const float s = alpha * nrm[e];
    const float4 v = *reinterpret_cast<const float4*>(src + (size_t)c * OUTC + cg);
    float* d = dst + (size_t)r * OUTC + cg;
    atomicAdd(d + 0, s * v.x);
    atomicAdd(d + 1, s * v.y);
    atomicAdd(d + 2, s * v.z);
    atomicAdd(d + 3, s * v.w);
}

// ----------------- Scalar / elementwise kernels ----------------------------
// coef[0..5]=softmax(theta), coef[6]=tanh(hp)
__global__ void coef_kernel(float* __restrict__ coef, const float* __restrict__ theta,
                            const float* __restrict__ hp) {
    if (threadIdx.x == 0 && blockIdx.x == 0) {
        float m = theta[0];
        for (int k = 1; k < 6; ++k) m = fmaxf(m, theta[k]);
        float s = 0.0f, e[6];
        for (int k = 0; k < 6; ++k) { e[k] = __expf(theta[k] - m); s += e[k]; }
        for (int k = 0; k < 6; ++k) coef[k] = e[k] / s;
        coef[6] = tanhf(hp[0]);
    }
}

__global__ void ew_zero4(float4* __restrict__ d, int n4) {
    int i = blockIdx.x * blockDim.x + threadIdx.x;
    if (i < n4) d[i] = make_float4(0.f, 0.f, 0.f, 0.f);
}

__global__ void ew_scale4(float4* __restrict__ d, const float4* __restrict__ s,
                          float a, int n4) {
    int i = blockIdx.x * blockDim.x + threadIdx.x;
    if (i < n4) {
        float4 v = s[i];
        d[i] = make_float4(a * v.x, a * v.y, a * v.z, a * v.w);
    }
}

// d += (coef[i0] + (i1>=0 ? coef[i1] : 0)) * s
__global__ void ew_axpy_coef4(float4* __restrict__ d, const float4* __restrict__ s,
                              const float* __restrict__ coef, int i0, int i1, int n4) {
    int i = blockIdx.x * blockDim.x + threadIdx.x;
    if (i < n4) {
        float a = coef[i0] + (i1 >= 0 ? coef[i1] : 0.0f);
        float4 v = s[i], o = d[i];
        d[i] = make_float4(o.x + a * v.x, o.y + a * v.y, o.z + a * v.z, o.w + a * v.w);
    }
}

// ----------------- Host orchestration --------------------------------------
extern "C" void kernel_launch(void* const* d_in, const int* in_sizes, int n_in,
                              void* d_out, int out_size, void* d_ws, size_t ws_size,
                              hipStream_t stream) {
    const float* x     = (const float*)d_in[0];
    const int*   ei    = (const int*)d_in[1];   // [2,E] int32 (JAX x64 disabled)
    const float* ew    = (const float*)d_in[2];
    const float* Win   = (const float*)d_in[3];
    const float* theta = (const float*)d_in[4];
    const float* Wout  = (const float*)d_in[5];
    const float* bout  = (const float*)d_in[6];
    const float* Wg    = (const float*)d_in[7];
    const float* bg    = (const float*)d_in[8];
    const float* hp    = (const float*)d_in[9];

    const int E = in_sizes[2];
    const int N = in_sizes[0] / INC;
    const size_t NC = (size_t)N * OUTC;

    const int* row = ei;
    const int* col = ei + E;

    // workspace carve-up (floats)
    float* w    = (float*)d_ws;
    size_t off  = 0;
    float* deg  = w + off; off += ((size_t)N + 255) & ~(size_t)255;
    float* nrm  = w + off; off += ((size_t)E + 255) & ~(size_t)255;
    float* h0   = w + off; off += NC;      // T0
    float* Ta   = w + off; off += NC;
    float* Tb   = w + off; off += NC;
    float* Tc   = w + off; off += NC;
    float* Hacc = w + off; off += NC;
    float* H    = w + off; off += NC;
    float* coef = w + off; off += 8;

    const int n4      = (int)(NC / 4);
    const int gEw     = (n4 + 255) / 256;          // elementwise grid over N*64
    const int gEdge   = (E + 255) / 256;
    const int gSpmm   = (E * 16 + 255) / 256;
    const int gTiles  = N / 16;                    // WMMA grid (4 waves/block)
    const int gDeg4   = ((N + 3) / 4 + 255) / 256;

    // 1. degree + normalization
    ew_zero4<<<gDeg4, 256, 0, stream>>>((float4*)deg, (N + 3) / 4);
    deg_kernel<<<gEdge, 256, 0, stream>>>(deg, ew, row, E);
    norm_kernel<<<gEdge, 256, 0, stream>>>(nrm, deg, ew, row, col, E);

    // 2. h0 = x @ W_in  (WMMA fp32)
    gemm_in_kernel<<<gTiles, 128, 0, stream>>>(x, Win, h0);

    // 3. coefficients on device
    coef_kernel<<<1, 32, 0, stream>>>(coef, theta, hp);

    // 4. Hacc = (w0 + tanh(hp)) * h0
    ew_zero4<<<gEw, 256, 0, stream>>>((float4*)Hacc, n4);
    ew_axpy_coef4<<<gEw, 256, 0, stream>>>((float4*)Hacc, (const float4*)h0, coef, 0, 6, n4);

    // 5. T1 = -spmm(h0);  Hacc += (w1 + tanh(hp)) * T1
    ew_zero4<<<gEw, 256, 0, stream>>>((float4*)Ta, n4);
    spmm_scatter_kernel<<<gSpmm, 256, 0, stream>>>(Ta, h0, nrm, row, col, -1.0f, E);
    ew_axpy_coef4<<<gEw, 256, 0, stream>>>((float4*)Hacc, (const float4*)Ta, coef, 1, 6, n4);

    // 6. Chebyshev recursion T_k = -2*spmm(T_{k-1}) - T_{k-2}, k = 2..5
    float* Tpp = h0;  // T_{k-2}
    float* Tp  = Ta;  // T_{k-1}
    float* Tn  = Tb;  // T_k
    float* spare = Tc;
    for (int k = 2; k <= 5; ++k) {
        ew_scale4<<<gEw, 256, 0, stream>>>((float4*)Tn, (const float4*)Tpp, -1.0f, n4);
        spmm_scatter_kernel<<<gSpmm, 256, 0, stream>>>(Tn, Tp, nrm, row, col, -2.0f, E);
        ew_axpy_coef4<<<gEw, 256, 0, stream>>>((float4*)Hacc, (const float4*)Tn, coef, k, -1, n4);
        // rotate (never overwrite h0, it is needed for the gate)
        float* newSpare = (Tpp == h0) ? spare : Tpp;
        Tpp = Tp; Tp = Tn; Tn = newSpare; spare = newSpare;
    }

    // 7. H = Hacc @ W_out + b_out  (WMMA fp32)
    gemm_out_kernel<<<gTiles, 128, 0, stream>>>(Hacc, Wout, bout, H);

    // 8. gate + blend -> d_out  (two WMMA accumulations)
    gate_kernel<<<gTiles, 128, 0, stream>>>(H, h0, Wg, bg, (float*)d_out);
}